// VariantEmbedder_61572651155962
// MI455X (gfx1250) — compile-verified
//
#include <hip/hip_runtime.h>
#include <cstdint>
#include <math.h>

#define DIM 64
#define TILE_ROWS 8          // rows staged per async tile (per wave)
#define CPS_PER_TILE 4       // async-copy instructions per tile (2 rows each)
#define MAX_NC 64            // >= n_clusters (50)
#define FALLBACK_SEGS 100000

// b128 payload type the gfx1250 async builtins expect (per hipcc diagnostic:
// parameter is '__attribute__((vector_size(4*sizeof(int)))) int __device__*',
// i.e. int4 in addrspace(1); LDS side is the addrspace(3) analogue).
typedef int v4i_ __attribute__((vector_size(16)));

// ---------------------------------------------------------------------------
// CDNA5 async global->LDS copy (16B/lane) + ASYNCcnt waits.
// ---------------------------------------------------------------------------
#if defined(__gfx1250__)
  #if __has_builtin(__builtin_amdgcn_global_load_async_to_lds_b128)
    #define ASYNC_CP16(gsrc, ldst)                                          \
      __builtin_amdgcn_global_load_async_to_lds_b128(                       \
          (__attribute__((address_space(1))) v4i_*)(gsrc),                  \
          (__attribute__((address_space(3))) v4i_*)(ldst), 0, 0)
  #else
    #define ASYNC_CP16(gsrc, ldst)                                          \
      asm volatile("global_load_async_to_lds_b128 %0, %1, off"              \
                   :                                                        \
                   : "v"((__attribute__((address_space(3))) v4i_*)(ldst)),  \
                     "v"((__attribute__((address_space(1))) v4i_*)(gsrc))   \
                   : "memory")
  #endif
  #if __has_builtin(__builtin_amdgcn_s_wait_asynccnt)
    #define WAIT_ASYNC(n) __builtin_amdgcn_s_wait_asynccnt(n)
  #else
    #define WAIT_ASYNC(n) asm volatile("s_wait_asynccnt %0" ::"i"(n) : "memory")
  #endif
#else
  // host-compile stub (never executed on device)
  #define ASYNC_CP16(gsrc, ldst) (*(float4*)(ldst) = *(const float4*)(gsrc))
  #define WAIT_ASYNC(n) ((void)0)
#endif

// scratch fallback if the harness workspace is too small
__device__ float g_raw_fallback[(size_t)FALLBACK_SEGS * DIM];

// ---------------------------------------------------------------------------
// Phase 1: per-segment sum of contiguous rows. One wave32 per segment.
// Lane l: dim4 = l&15 (16B column), roff = l>>4 (row parity). A wave covers
// 2 rows x 64 dims per copy instruction; 8-row tiles are double-buffered in
// LDS via the async copy engine.
// ---------------------------------------------------------------------------
__global__ __launch_bounds__(32)
void seg_sum_kernel(const float* __restrict__ emb,
                    const void* __restrict__ indptr_raw,
                    float* __restrict__ raw_out,
                    int n_seg)
{
  __shared__ __align__(16) float tile[2][TILE_ROWS * DIM];

  const int s    = blockIdx.x;
  const int lane = threadIdx.x;
  const int dim4 = lane & 15;
  const int roff = lane >> 4;
  if (s >= n_seg) return;

  float* raw = raw_out ? raw_out : g_raw_fallback;

  // dtype sniff: int64 indptr has zero high-words for the first (small)
  // entries; int32 indptr would need >=7 leading breakpoints == 0 (p ~ 0).
  const int* w32 = (const int*)indptr_raw;
  const bool is64 = (w32[1] == 0) & (w32[3] == 0) & (w32[5] == 0) & (w32[7] == 0);
  long long start, end;
  if (is64) {
    const long long* p = (const long long*)indptr_raw;
    start = p[s];
    end   = p[s + 1];
  } else {
    start = (long long)w32[s];
    end   = (long long)w32[s + 1];
  }

  float4 acc = make_float4(0.f, 0.f, 0.f, 0.f);
  const long long nrows  = end - start;
  const int       ntiles = (int)(nrows >> 3);          // full 8-row tiles
  const float*    gcol   = emb + (size_t)dim4 * 4;     // this lane's 16B column

  // stage one 8-row tile into LDS buffer b (4 async B128 ops per lane)
  auto stage = [&](long long row0, int b) {
    #pragma unroll
    for (int k = 0; k < CPS_PER_TILE; ++k) {
      const int r = roff + 2 * k;
      ASYNC_CP16(gcol + (size_t)(row0 + r) * DIM,
                 &tile[b][r * DIM + dim4 * 4]);
    }
  };

  if (ntiles > 0) stage(start, 0);
  for (int t = 0; t < ntiles; ++t) {
    if (t + 1 < ntiles) {
      stage(start + (long long)(t + 1) * TILE_ROWS, (t + 1) & 1);
      WAIT_ASYNC(CPS_PER_TILE);   // tile t landed; tile t+1 still in flight
    } else {
      WAIT_ASYNC(0);
    }
    const float* lbuf = &tile[t & 1][0];
    #pragma unroll
    for (int k = 0; k < CPS_PER_TILE; ++k) {
      const int r = roff + 2 * k;
      const float4 v = *(const float4*)&lbuf[r * DIM + dim4 * 4];
      acc.x += v.x; acc.y += v.y; acc.z += v.z; acc.w += v.w;
    }
  }

  // remainder rows (< TILE_ROWS): direct 128b global loads
  for (long long r = start + (long long)ntiles * TILE_ROWS + roff; r < end; r += 2) {
    const float4 v = *(const float4*)(gcol + (size_t)r * DIM);
    acc.x += v.x; acc.y += v.y; acc.z += v.z; acc.w += v.w;
  }

  // fold the two row-parity partials (lane l += lane l+16), wave32 shuffle
  acc.x += __shfl_down(acc.x, 16);
  acc.y += __shfl_down(acc.y, 16);
  acc.z += __shfl_down(acc.z, 16);
  acc.w += __shfl_down(acc.w, 16);

  if (lane < 16) {
    *(float4*)(raw + (size_t)s * DIM + dim4 * 4) = acc;
  }
}

// ---------------------------------------------------------------------------
// Phase 2: normalize + log1p, then mean/std(ddof=1) across clusters, emit
// [ve, rel]. One 64-thread block per variant; thread d owns dim d. All 50
// cluster values are kept in LDS for an accurate two-pass moment computation.
// Intermediate (25.6 MB) is expected to be L2-resident (192 MB L2).
// ---------------------------------------------------------------------------
__global__ __launch_bounds__(64)
void finalize_kernel(const float* __restrict__ raw_in,
                     const float* __restrict__ lib,
                     float* __restrict__ out,
                     int nc, int nv)
{
  __shared__ float vals[MAX_NC * DIM];
  __shared__ float slib[MAX_NC];

  const float* raw = raw_in ? raw_in : g_raw_fallback;
  const int v = blockIdx.x;
  const int d = threadIdx.x;

  if (d < nc) slib[d] = lib[d];
  __syncthreads();

  float sum = 0.f;
  for (int c = 0; c < nc; ++c) {
    const float rr = raw[((size_t)(c * nv + v)) * DIM + d];
    const float ve = log1pf(rr / slib[c]) - 2.0f;
    vals[c * DIM + d] = ve;
    sum += ve;
  }
  const float mean = sum / (float)nc;

  float ss = 0.f;
  for (int c = 0; c < nc; ++c) {
    const float t = vals[c * DIM + d] - mean;
    ss += t * t;
  }
  const float stdv = sqrtf(ss / (float)(nc - 1));
  const float inv  = 1.0f / (stdv + 1e-5f);

  for (int c = 0; c < nc; ++c) {
    const float  ve   = vals[c * DIM + d];
    const size_t base = ((size_t)(c * nv + v)) * (2 * DIM);
    out[base + d]       = ve;
    out[base + DIM + d] = (ve - mean) * inv;
  }
}

// ---------------------------------------------------------------------------
extern "C" void kernel_launch(void* const* d_in, const int* in_sizes, int n_in,
                              void* d_out, int out_size, void* d_ws, size_t ws_size,
                              hipStream_t stream) {
  const float* emb    = (const float*)d_in[0];   // (n_cuts, 64) f32
  const float* lib    = (const float*)d_in[1];   // (n_clusters,) f32
  const void*  indptr = d_in[2];                 // (n_seg+1,) int32 or int64

  const int nc    = in_sizes[1];                 // 50
  const int n_seg = in_sizes[2] - 1;             // 100000
  const int nv    = n_seg / nc;                  // 2000
  float* out = (float*)d_out;

  const size_t need = (size_t)n_seg * DIM * sizeof(float);
  float* raw = nullptr;
  if (ws_size >= need) {
    raw = (float*)d_ws;                          // normal path
  } else if ((size_t)n_seg * DIM > (size_t)FALLBACK_SEGS * DIM) {
    raw = (float*)d_ws;                          // fallback too small; best effort
  }                                              // else raw==nullptr -> device global

  seg_sum_kernel<<<n_seg, 32, 0, stream>>>(emb, indptr, raw, n_seg);
  finalize_kernel<<<nv, 64, 0, stream>>>(raw, lib, out, nc, nv);
}